// GDN_51110110822634
// MI455X (gfx1250) — compile-verified
//
#include <hip/hip_runtime.h>
#include <hip/hip_bf16.h>

typedef float v2f __attribute__((ext_vector_type(2)));
typedef float v8f __attribute__((ext_vector_type(8)));

#define FEATS   512
#define NW      5
#define LOG2E   1.4426950408889634f

// ---------------------------------------------------------------------------
// Kernel 1: x[f] = sum_w data[w][f]  (att == softmax over batch dim of size 1
// == ones, so the attention MLP contributes nothing).  Also reduce x_max /
// x_min so the GAT softmax max can be computed analytically (leaky is
// monotone, logits are affine in x[j]).
// ---------------------------------------------------------------------------
__global__ void gdn_reduce_x(const float* __restrict__ data,
                             float* __restrict__ wsx) {
    __shared__ float smax[FEATS];
    __shared__ float smin[FEATS];
    const int f = threadIdx.x;           // blockDim.x == 512
    float x = 0.f;
#pragma unroll
    for (int w = 0; w < NW; ++w) x += data[w * FEATS + f];
    wsx[f]  = x;
    smax[f] = x;
    smin[f] = x;
    __syncthreads();
    for (int off = FEATS / 2; off > 0; off >>= 1) {
        if (f < off) {
            smax[f] = fmaxf(smax[f], smax[f + off]);
            smin[f] = fminf(smin[f], smin[f + off]);
        }
        __syncthreads();
    }
    if (f == 0) {
        wsx[FEATS]     = smax[0];
        wsx[FEATS + 1] = smin[0];
    }
}

// ---------------------------------------------------------------------------
// Kernel 2: the 512^3 GAT softmax-weighted reduction (dominant cost).
// One thread per (dst i, head h).  feat_r[i,h] = gat_w[h] *
//   sum_j softmax_j(leaky(x[i]*u + x[j]*v, 0.2)) * x[j]  + gat_bias[h]
// with u = gat_w*a_dst, v = gat_w*a_src.  Softmax max is analytic:
//   max_j t = p + (v>=0 ? v*x_max : v*x_min), leaky monotone -> emax known.
// Single pass, one v_exp per j.  x[] broadcast from LDS.
// ---------------------------------------------------------------------------
__global__ void gdn_gat(const float* __restrict__ wsx,
                        const float* __restrict__ gat_w,
                        const float* __restrict__ a_src,
                        const float* __restrict__ a_dst,
                        const float* __restrict__ gat_bias,
                        float* __restrict__ featr) {
    __shared__ float lx[FEATS];
    const int t = threadIdx.x;           // blockDim.x == 256
    lx[t]       = wsx[t];
    lx[t + 256] = wsx[t + 256];
    __syncthreads();

    const int gid = blockIdx.x * 256 + t;
    const int i   = gid >> 9;            // dst node  (wave-uniform: 512 h per i)
    const int h   = gid & (FEATS - 1);   // head      (consecutive -> coalesced)

    const float gw = gat_w[h];
    const float u  = gw * a_dst[h];
    const float v  = gw * a_src[h];
    const float xi = lx[i];

    const float xmax = wsx[FEATS];
    const float xmin = wsx[FEATS + 1];

    const float p    = xi * u;
    const float tmax = p + (v >= 0.f ? v * xmax : v * xmin);
    const float emax = (tmax >= 0.f) ? tmax : 0.2f * tmax;

    float s = 0.f, acc = 0.f;
    for (int j = 0; j < FEATS; j += 4) {
        const float4 xj = *(const float4*)&lx[j];   // LDS broadcast b128
        const float xs[4] = {xj.x, xj.y, xj.z, xj.w};
#pragma unroll
        for (int q = 0; q < 4; ++q) {
            const float tt  = fmaf(xs[q], v, p);
            const float e   = (tt >= 0.f) ? tt : 0.2f * tt;
            const float wgt = __builtin_amdgcn_exp2f((e - emax) * LOG2E);
            s   += wgt;
            acc  = fmaf(wgt, xs[q], acc);
        }
    }
    featr[i * FEATS + h] = gw * (acc / s) + gat_bias[h];
}

// ---------------------------------------------------------------------------
// Kernel 3: y1 = feat_r(512x512) @ F1^T(512x16) + fb1  via
// V_WMMA_F32_16X16X4_F32 (fp32 matrix path, no precision loss), then the
// fused tiny second layer  out[i,w] = sigmoid(y1[i,:] . F2[w,:] + fb2[w]).
// One wave (32 lanes) per 16-row tile; K=512 walked in steps of 4.
// A 16x4 f32 layout: lanes 0-15 hold K={kb,kb+1}, lanes 16-31 K={kb+2,kb+3};
// B 4x16 mirrored (lane%16 = column n).
// ---------------------------------------------------------------------------
__global__ void gdn_fcn(const float* __restrict__ featr,
                        const float* __restrict__ F1,
                        const float* __restrict__ fb1,
                        const float* __restrict__ F2,
                        const float* __restrict__ fb2,
                        float* __restrict__ out) {
    __shared__ float tile[16][17];
    const int lane = threadIdx.x;        // blockDim.x == 32 (one wave)
    const int m    = lane & 15;
    const int half = lane >> 4;
    const int i0   = blockIdx.x * 16;

    const float* arow = featr + (size_t)(i0 + m) * FEATS;  // A row (M = m)
    const float* brow = F1    + (size_t)m * FEATS;         // B col n = m (F1^T)

    v8f c = {0.f, 0.f, 0.f, 0.f, 0.f, 0.f, 0.f, 0.f};
    for (int kb = 0; kb < FEATS; kb += 4) {
        const int kA = kb + 2 * half;                      // even -> b64 aligned
        const v2f a = *(const v2f*)&arow[kA];
        const v2f b = *(const v2f*)&brow[kA];
        c = __builtin_amdgcn_wmma_f32_16x16x4_f32(
                /*neg_a=*/false, a, /*neg_b=*/false, b,
                /*c_mod=*/(short)0, c, /*reuse_a=*/false, /*reuse_b=*/false);
    }

    // D layout: lane L -> column n = L%16; VGPR r -> row (r + 8*(L/16)).
    // Bias fb1 indexed by column n = m; leaky(.,1.0) is identity.
    const float bias = fb1[m];
#pragma unroll
    for (int r = 0; r < 8; ++r)
        tile[r + 8 * half][m] = c[r] + bias;
    __syncthreads();

    // Second layer (16 -> 5) + sigmoid, 80 outputs per tile.
    for (int idx = lane; idx < 16 * NW; idx += 32) {
        const int row = idx / NW;
        const int w   = idx % NW;
        float z = fb2[w];
#pragma unroll
        for (int k = 0; k < 16; ++k)
            z = fmaf(tile[row][k], F2[w * 16 + k], z);
        const float ez = __builtin_amdgcn_exp2f(-z * LOG2E);
        out[(i0 + row) * NW + w] = __builtin_amdgcn_rcpf(1.f + ez);
    }
}

// ---------------------------------------------------------------------------
// Launch.  Workspace: ws[0..511] = x, ws[512]=xmax, ws[513]=xmin,
// feat_r (512*512 f32, 1 MB) at float offset 1024 (4 KB aligned).
// Inputs (setup_inputs order): 0 data, 1 W1, 2 b1, 3 W2, 4 b2, 5 W3, 6 b3,
// 7 gat_w, 8 a_src, 9 a_dst, 10 gat_bias, 11 F1, 12 fb1, 13 F2, 14 fb2.
// W1..b3 are mathematically dead (softmax over batch dim of size 1).
// ---------------------------------------------------------------------------
extern "C" void kernel_launch(void* const* d_in, const int* in_sizes, int n_in,
                              void* d_out, int out_size, void* d_ws, size_t ws_size,
                              hipStream_t stream) {
    const float* data     = (const float*)d_in[0];
    const float* gat_w    = (const float*)d_in[7];
    const float* a_src    = (const float*)d_in[8];
    const float* a_dst    = (const float*)d_in[9];
    const float* gat_bias = (const float*)d_in[10];
    const float* F1       = (const float*)d_in[11];
    const float* fb1      = (const float*)d_in[12];
    const float* F2       = (const float*)d_in[13];
    const float* fb2      = (const float*)d_in[14];
    float*       out      = (float*)d_out;

    float* wsf   = (float*)d_ws;
    float* wsx   = wsf;            // 514 floats
    float* featr = wsf + 1024;     // 512*512 floats

    gdn_reduce_x<<<1, FEATS, 0, stream>>>(data, wsx);
    gdn_gat<<<(FEATS * FEATS) / 256, 256, 0, stream>>>(
        wsx, gat_w, a_src, a_dst, gat_bias, featr);
    gdn_fcn<<<FEATS / 16, 32, 0, stream>>>(featr, F1, fb1, F2, fb2, out);
}